// PSRoiAlignPooling_12421045420776
// MI455X (gfx1250) — compile-verified
//
#include <hip/hip_runtime.h>
#include <hip/hip_bf16.h>
#include <stdint.h>

// ---------------------------------------------------------------------------
// PS-RoI Align for MI455X (gfx1250).
// - Tensor Data Mover stages each bin's 20x20x32 f32 slice (51.2 KB strided
//   2D tile) into LDS once per block (tensor_load_to_lds + s_wait_tensorcnt).
// - Each wave32 processes 4 ROIs: lane group (lane>>3) = ROI, lane&7 = float4
//   channel slice -> ds_load_b128 taps engage all 64 LDS banks, and stores
//   are global_store_b128 (4x fewer ds/vmem issue slots than the b32 form).
// ---------------------------------------------------------------------------

#define POOLN       7
#define TOTAL_BINS  49
#define ALPHA       32
#define NUM_ROIS    8192
#define IMG_H       20
#define IMG_W       20
#define IMG_C       (TOTAL_BINS * ALPHA)     // 1568 floats per texel
#define ROIS_PER_BLOCK 256
#define THREADS     256                      // 8 wave32s
#define NWAVES      (THREADS / 32)

typedef unsigned int v4u __attribute__((ext_vector_type(4)));
typedef int          v8i __attribute__((ext_vector_type(8)));
typedef int          v4i __attribute__((ext_vector_type(4)));

__global__ __launch_bounds__(THREADS)
void psroi_align_kernel(const float* __restrict__ img,
                        const float* __restrict__ rois,
                        float* __restrict__ out) {
    __shared__ float tile[IMG_H * IMG_W * ALPHA];   // 51200 bytes

    const int bin  = blockIdx.x;            // 0..48
    const int lane = threadIdx.x & 31;
    const int wave = threadIdx.x >> 5;
    const int g    = lane >> 3;             // ROI sub-group 0..3
    const int cg   = lane & 7;              // float4 channel slot 0..7

    // ---- Stage bin slice into LDS via Tensor Data Mover (wave 0 issues) ---
#if defined(__has_builtin) && __has_builtin(__builtin_amdgcn_tensor_load_to_lds)
    if (threadIdx.x < 32) {
        uint64_t gaddr = (uint64_t)(uintptr_t)img + (uint64_t)bin * (ALPHA * 4u);
        uint32_t ldsa  = (uint32_t)(uintptr_t)&tile[0];   // low 32b of flat = LDS byte addr

        // D# group 0 (128b): count=1 (valid), lds_addr, global_addr[56:0], type=2
        v4u g0;
        g0.x = 1u;                                        // count=1, user mode
        g0.y = ldsa;                                      // lds_addr
        g0.z = (uint32_t)gaddr;                           // global_addr[31:0]
        g0.w = 0x80000000u | (uint32_t)((gaddr >> 32) & 0x01FFFFFFu); // addr[56:32] | type=2

        // D# group 1 (256b): data_size=4B, tensor 32 x 400, tile 32 x 400,
        // row stride 1568 floats (6272 B). No multicast, no pad, no iterate.
        v8i g1;
        g1[0] = (2 << 16);          // data_size = 2 (4 bytes)
        g1[1] = (32 << 16);         // tensor_dim0 = 32 (bits 79:48, low half)
        g1[2] = (400 << 16);        // tensor_dim0 hi16 = 0 | tensor_dim1 = 400 (low16)
        g1[3] = (32 << 16);         // tensor_dim1 hi16 = 0 | tile_dim0 = 32
        g1[4] = 400;                // tile_dim1 = 400, tile_dim2 = 0
        g1[5] = 1568;               // tensor_dim0_stride[31:0]
        g1[6] = 0;                  // stride hi | tensor_dim1_stride lo (unused, 2D)
        g1[7] = 0;

        v4i z4 = {0, 0, 0, 0};
#if __clang_major__ >= 23
        v8i z8 = {0, 0, 0, 0, 0, 0, 0, 0};
        __builtin_amdgcn_tensor_load_to_lds(g0, g1, z4, z4, z8, 0);
#else
        __builtin_amdgcn_tensor_load_to_lds(g0, g1, z4, z4, 0);
#endif
        __builtin_amdgcn_s_wait_tensorcnt(0);
    }
#else
    // Fallback: cooperative copy (should not be needed on gfx1250 toolchains)
    for (int e = threadIdx.x; e < IMG_H * IMG_W * ALPHA; e += THREADS) {
        int texel = e >> 5, a = e & 31;
        tile[e] = img[(size_t)texel * IMG_C + bin * ALPHA + a];
    }
#endif
    __syncthreads();

    // ---- Per-ROI bilinear pooling (4 ROIs per wave, float4 channels) ------
    const float4* rois4   = (const float4*)rois;
    const float4* t4      = (const float4*)tile;          // (y*20+x)*8 + cg
    float4*       out4    = (float4*)out;
    const int     roiBase = blockIdx.y * ROIS_PER_BLOCK;
    const float   px      = (float)(bin / POOLN);   // bin_x
    const float   py      = (float)(bin % POOLN);   // bin_y
    const float   sc      = (float)(IMG_H - 1) / (float)IMG_H;  // 19/20

    #pragma unroll 2
    for (int it = 0; it < ROIS_PER_BLOCK / (NWAVES * 4); ++it) {
        const int ri = roiBase + it * (NWAVES * 4) + wave * 4 + g;
        float4 roi = rois4[ri];                     // x, y, w, h
        float stepx = roi.z * (1.0f / POOLN);
        float stepy = roi.w * (1.0f / POOLN);

        float xA = (roi.x + px * stepx) * sc;
        float xB = (roi.x + (px + 1.0f) * stepx) * sc;
        float yA = (roi.y + py * stepy) * sc;
        float yB = (roi.y + (py + 1.0f) * stepy) * sc;

        // Per-axis floor indices (clamped) and lerp weights
        float ffxA = floorf(xA), ffxB = floorf(xB);
        float ffyA = floorf(yA), ffyB = floorf(yB);
        float wxA = xA - ffxA, wxB = xB - ffxB;
        float wyA = yA - ffyA, wyB = yB - ffyB;

        int xi[4], yi[4];
        xi[0] = min(max((int)ffxA, 0), IMG_W - 1); xi[1] = min(xi[0] + 1, IMG_W - 1);
        xi[2] = min(max((int)ffxB, 0), IMG_W - 1); xi[3] = min(xi[2] + 1, IMG_W - 1);
        yi[0] = min(max((int)ffyA, 0), IMG_H - 1); yi[1] = min(yi[0] + 1, IMG_H - 1);
        yi[2] = min(max((int)ffyB, 0), IMG_H - 1); yi[3] = min(yi[2] + 1, IMG_H - 1);

        // Sample-point validity masks, x0.25 (mean over the fixed 2x2 grid)
        bool vxA = (xA >= 0.0f) && (xA <= (float)(IMG_W - 1));
        bool vxB = (xB >= 0.0f) && (xB <= (float)(IMG_W - 1));
        bool vyA = (yA >= 0.0f) && (yA <= (float)(IMG_H - 1));
        bool vyB = (yB >= 0.0f) && (yB <= (float)(IMG_H - 1));
        float mm[2][2];
        mm[0][0] = (vyA && vxA) ? 0.25f : 0.0f;
        mm[0][1] = (vyA && vxB) ? 0.25f : 0.0f;
        mm[1][0] = (vyB && vxA) ? 0.25f : 0.0f;
        mm[1][1] = (vyB && vxB) ? 0.25f : 0.0f;

        float yw[4] = { 1.0f - wyA, wyA, 1.0f - wyB, wyB };
        float xw[4] = { 1.0f - wxA, wxA, 1.0f - wxB, wxB };

        // Accumulate the 16 texel taps: weight = mask * yw[i] * xw[j]
        float4 acc; acc.x = acc.y = acc.z = acc.w = 0.0f;
        #pragma unroll
        for (int i = 0; i < 4; ++i) {
            const int rowBase = yi[i] * IMG_W;
            const float wyi = yw[i];
            #pragma unroll
            for (int j = 0; j < 4; ++j) {
                float w = mm[i >> 1][j >> 1] * wyi * xw[j];
                float4 v = t4[(rowBase + xi[j]) * (ALPHA / 4) + cg];  // ds_load_b128
                acc.x = fmaf(w, v.x, acc.x);
                acc.y = fmaf(w, v.y, acc.y);
                acc.z = fmaf(w, v.z, acc.z);
                acc.w = fmaf(w, v.w, acc.w);
            }
        }

        out4[((size_t)ri * TOTAL_BINS + bin) * (ALPHA / 4) + cg] = acc;  // global_store_b128
    }
}

extern "C" void kernel_launch(void* const* d_in, const int* in_sizes, int n_in,
                              void* d_out, int out_size, void* d_ws, size_t ws_size,
                              hipStream_t stream) {
    const float* img  = (const float*)d_in[0];   // (1, 20, 20, 1568) f32
    const float* rois = (const float*)d_in[1];   // (1, 8192, 4) f32
    float*       out  = (float*)d_out;           // (1, 8192, 7, 7, 32) f32

    dim3 grid(TOTAL_BINS, NUM_ROIS / ROIS_PER_BLOCK);   // (49, 32)
    psroi_align_kernel<<<grid, THREADS, 0, stream>>>(img, rois, out);
}